// Memory_70007966925506
// MI455X (gfx1250) — compile-verified
//
#include <hip/hip_runtime.h>
#include <math.h>

#define BB 16
#define AA 16384
#define KK 128
#define MM 524288
#define TOPK 256
#define NCAND 4096
#define NBINS 2048

typedef __attribute__((ext_vector_type(2))) float v2f;
typedef __attribute__((ext_vector_type(4))) float v4f;
typedef __attribute__((ext_vector_type(8))) float v8f;

// ---- output layout (floats, concatenated in reference return order) ----
#define OFF_Y       0L
#define OFF_YHAT    16L
#define OFF_SOFT    32L
#define OFF_CLS     4128L
#define OFF_LOC     4129L
#define OFF_QUERY   4130L
#define OFF_KEYS    6178L
#define OFF_VALUES  67115042L
#define OFF_AGE     67639330L

// ---- workspace layout (byte offsets) ----
#define WS_SAMPLES  0        // 16*128 f32
#define WS_NUMPOS   8192     // 16 f32
#define WS_LOCSUM   8256     // 16 f32
#define WS_Y        8320     // 16 i32
#define WS_QUERY    8384     // 16*128 f32
#define WS_RESULT   16576    // 16 i32
#define WS_YHATIDX  16640    // 16 i32
#define WS_RANK     16704    // 16 i32
#define WS_OLDEST   16768    // 16 i32
#define WS_COSSIM   16896    // 16*256 f32
#define WS_TOPKIDX  33280    // 16*256 i32
#define WS_SCORES   65536    // 16*MM f32 (32 MB)

__global__ void init_ws(float* samples, float* numpos, float* locsum, int* y) {
  int i = blockIdx.x * blockDim.x + threadIdx.x;
  if (i < BB * KK) samples[i] = 0.f;
  if (i < BB) { numpos[i] = 0.f; locsum[i] = 0.f; y[i] = 0; }
}

// --- Phase 1: per-batch anchor reductions (streams cls_preds once) ---
__global__ __launch_bounds__(128) void reduce_anchors(
    const float* __restrict__ loc_preds, const float* __restrict__ loc_targets,
    const float* __restrict__ cls_preds, const int* __restrict__ cls_targets,
    float* ws_samples, float* ws_numpos, float* ws_locsum, int* ws_y) {
  __shared__ float spos[128];
  __shared__ float red[128];
  __shared__ int ired[128];
  int t = threadIdx.x;
  int b = blockIdx.y;
  int a0 = blockIdx.x * 128;
  int a = a0 + t;

  int tgt = cls_targets[(long)b * AA + a];
  float p = (tgt > 0) ? 1.f : 0.f;
  spos[t] = p;
  float locs = 0.f;
  if (p > 0.f) {
    float4 lp = ((const float4*)loc_preds)[(long)b * AA + a];
    float4 lt = ((const float4*)loc_targets)[(long)b * AA + a];
    float d0 = lp.x - lt.x, d1 = lp.y - lt.y, d2 = lp.z - lt.z, d3 = lp.w - lt.w;
    float dd[4] = {d0, d1, d2, d3};
#pragma unroll
    for (int i = 0; i < 4; ++i) {
      float ad = fabsf(dd[i]);
      locs += (ad < 1.f) ? 0.5f * dd[i] * dd[i] : (ad - 0.5f);
    }
  }
  __syncthreads();

  // thread t owns class-dim k=t; coalesced stream over 128 anchors
  float acc = 0.f;
  const float* cp = cls_preds + ((long)b * AA + a0) * KK + t;
  for (int al = 0; al < 128; ++al) acc += spos[al] * cp[(long)al * KK];
  atomicAdd(&ws_samples[b * KK + t], acc);

  red[t] = p; __syncthreads();
  for (int s = 64; s > 0; s >>= 1) { if (t < s) red[t] += red[t + s]; __syncthreads(); }
  if (t == 0) atomicAdd(&ws_numpos[b], red[0]);
  __syncthreads();
  red[t] = locs; __syncthreads();
  for (int s = 64; s > 0; s >>= 1) { if (t < s) red[t] += red[t + s]; __syncthreads(); }
  if (t == 0) atomicAdd(&ws_locsum[b], red[0]);
  __syncthreads();
  ired[t] = tgt; __syncthreads();
  for (int s = 64; s > 0; s >>= 1) { if (t < s) ired[t] = max(ired[t], ired[t + s]); __syncthreads(); }
  if (t == 0) atomicMax(&ws_y[b], ired[0]);
}

// --- Phase 2: query = l2norm(samples / num_pos) ---
__global__ __launch_bounds__(128) void compute_query(
    const float* ws_samples, const float* ws_numpos, float* ws_query, float* out_query) {
  __shared__ float red[128];
  int b = blockIdx.x, t = threadIdx.x;
  float s = ws_samples[b * KK + t] / ws_numpos[b];
  red[t] = s * s; __syncthreads();
  for (int st = 64; st > 0; st >>= 1) { if (t < st) red[t] += red[t + st]; __syncthreads(); }
  float nrm = fmaxf(sqrtf(red[0]), 1e-12f);
  float q = s / nrm;
  ws_query[b * KK + t] = q;
  out_query[b * KK + t] = q;
}

// --- Phase 3: scores[16 x M] = query(16x128) @ keys.T, via V_WMMA_F32_16X16X4_F32 ---
// Block = 128 threads = 4 waves; each wave computes a 16(batch) x 16(key) f32 tile.
__global__ __launch_bounds__(128) void scores_wmma(
    const float* __restrict__ keys, const float* __restrict__ query,
    float* __restrict__ scores) {
  __shared__ float sq[BB * KK];          // 8 KB  query, staged once
  __shared__ float sk[4][16 * KK];       // 32 KB key tiles, one per wave
  int t = threadIdx.x;
  int wave = t >> 5, lane = t & 31;

  for (int i = t; i < BB * KK; i += 128) sq[i] = query[i];

  long keyBase = ((long)blockIdx.x * 4 + wave) * 16;
  // contiguous 16 key rows = 8 KB: fully coalesced b128 copy into LDS
  const v4f* src = (const v4f*)(keys + keyBase * KK);
  v4f* dst = (v4f*)&sk[wave][0];
#pragma unroll
  for (int i = 0; i < 16; ++i) dst[lane + 32 * i] = src[lane + 32 * i];
  __syncthreads();

  // ISA 16x4 f32 A-layout: lanes 0-15 M=0..15 K=k0+0/1, lanes 16-31 M=0..15 K=k0+2/3
  int col = lane & 15;
  int kh = (lane >> 4) * 2;
  const float* myq = &sq[col * KK];
  const float* myk = &sk[wave][col * KK];

  v8f c = {};
#pragma unroll
  for (int k0 = 0; k0 < KK; k0 += 4) {
    v2f a, bv;
    a.x = myq[k0 + kh];     a.y = myq[k0 + kh + 1];
    bv.x = myk[k0 + kh];    bv.y = myk[k0 + kh + 1];
    c = __builtin_amdgcn_wmma_f32_16x16x4_f32(
        /*neg_a=*/false, a, /*neg_b=*/false, bv,
        /*c_mod=*/(short)0, c, /*reuse_a=*/false, /*reuse_b=*/false);
  }

  // C/D layout: lane n(0-15): N=n, VGPR r: M=r; lanes 16-31: M=8+r
  int mb = (lane >> 4) * 8;
  long n = keyBase + col;
#pragma unroll
  for (int r = 0; r < 8; ++r) scores[(long)(mb + r) * MM + n] = c[r];
}

// --- Phase 4: exact-ish top-256 per row: histogram threshold + bitonic sort ---
__global__ __launch_bounds__(512) void topk_row(
    const float* __restrict__ scores, float* cossim, int* topkidx) {
  __shared__ unsigned hist[NBINS];
  __shared__ float cV[NCAND];
  __shared__ int cI[NCAND];
  __shared__ int s_thr, s_cnt;
  int b = blockIdx.x, t = threadIdx.x;
  const float* row = scores + (long)b * MM;

  for (int i = t; i < NBINS; i += 512) hist[i] = 0u;
  if (t == 0) s_cnt = 0;
  __syncthreads();

  for (long i = t; i < MM; i += 512) {
    unsigned u = __float_as_uint(row[i]);
    u = (u & 0x80000000u) ? ~u : (u | 0x80000000u);  // monotone map
    atomicAdd(&hist[u >> 21], 1u);
  }
  __syncthreads();
  if (t == 0) {
    unsigned cum = 0; int thr = 0;
    for (int bin = NBINS - 1; bin >= 0; --bin) {
      cum += hist[bin];
      if (cum >= TOPK) { thr = bin; break; }
    }
    s_thr = thr;
  }
  __syncthreads();
  int thr = s_thr;
  for (long i = t; i < MM; i += 512) {
    float v = row[i];
    unsigned u = __float_as_uint(v);
    u = (u & 0x80000000u) ? ~u : (u | 0x80000000u);
    if ((int)(u >> 21) >= thr) {
      int p = atomicAdd(&s_cnt, 1);
      if (p < NCAND) { cV[p] = v; cI[p] = (int)i; }
    }
  }
  __syncthreads();
  int cnt = min(s_cnt, NCAND);
  for (int i = t; i < NCAND; i += 512)
    if (i >= cnt) { cV[i] = -3.4e38f; cI[i] = 0; }
  __syncthreads();

  // bitonic sort NCAND entries descending by value (index payload)
  for (int k2 = 2; k2 <= NCAND; k2 <<= 1) {
    for (int j = k2 >> 1; j > 0; j >>= 1) {
      for (int i = t; i < NCAND; i += 512) {
        int ixj = i ^ j;
        if (ixj > i) {
          bool desc = ((i & k2) == 0);
          float va = cV[i], vb = cV[ixj];
          bool sw = desc ? (va < vb) : (va > vb);
          if (sw) {
            cV[i] = vb; cV[ixj] = va;
            int tmp = cI[i]; cI[i] = cI[ixj]; cI[ixj] = tmp;
          }
        }
      }
      __syncthreads();
    }
  }
  if (t < TOPK) {
    cossim[b * TOPK + t] = cV[t];
    topkidx[b * TOPK + t] = cI[t];
  }
}

// --- Phase 5: softmax, margin loss, loc loss, bookkeeping ---
__global__ __launch_bounds__(256) void finalize_kernel(
    const float* cossim, const int* topkidx, const int* __restrict__ values,
    const int* ws_y, const float* ws_numpos, const float* ws_locsum,
    int* ws_result, int* ws_yhatidx, int* ws_rank, float* out) {
  __shared__ float red[256];
  int t = threadIdx.x;
  float cls_acc = 0.f;
  for (int b = 0; b < BB; ++b) {
    float cv = cossim[b * TOPK + t];
    int idx = topkidx[b * TOPK + t];
    int y = ws_y[b];
    float mask = (values[idx] == y) ? 1.f : 0.f;
    float mx = cossim[b * TOPK];            // sorted: element 0 is the max
    float e = expf(cv - mx);
    __syncthreads();
    red[t] = e; __syncthreads();
    for (int s = 128; s > 0; s >>= 1) { if (t < s) red[t] += red[t + s]; __syncthreads(); }
    float denom = red[0]; __syncthreads();
    out[OFF_SOFT + b * TOPK + t] = e / denom;

    red[t] = cv * mask; __syncthreads();
    for (int s = 128; s > 0; s >>= 1) { if (t < s) red[t] = fmaxf(red[t], red[t + s]); __syncthreads(); }
    float posm = red[0]; __syncthreads();
    red[t] = cv * (1.f - mask); __syncthreads();
    for (int s = 128; s > 0; s >>= 1) { if (t < s) red[t] = fmaxf(red[t], red[t + s]); __syncthreads(); }
    float negm = red[0]; __syncthreads();
    red[t] = mask; __syncthreads();
    for (int s = 128; s > 0; s >>= 1) { if (t < s) red[t] += red[t + s]; __syncthreads(); }
    float cnt = red[0]; __syncthreads();

    if (t == 0) {
      float hp = (cnt > 0.f) ? 1.f : 0.f;
      cls_acc += fmaxf(negm - posm * hp + 0.1f, 0.f);
      int yhidx = topkidx[b * TOPK];
      int yhat = values[yhidx];
      out[OFF_Y + b] = (float)y;
      out[OFF_YHAT + b] = (float)yhat;
      ws_result[b] = (yhat == y) ? 1 : 0;
      ws_yhatidx[b] = yhidx;
    }
  }
  if (t == 0) {
    out[OFF_CLS] = cls_acc / (float)BB;
    float ls = 0.f, np = 0.f;
    for (int b = 0; b < BB; ++b) { ls += ws_locsum[b]; np += ws_numpos[b]; }
    out[OFF_LOC] = ls / np;
    int r = 0;
    for (int b = 0; b < BB; ++b) {
      if (!ws_result[b]) { ws_rank[b] = (r < BB - 1) ? r : BB - 1; ++r; }
      else ws_rank[b] = 0;
    }
  }
}

// --- Output passthrough copies ---
__global__ void copy_keys(const float* __restrict__ keys, float* __restrict__ out_keys) {
  const float2* src = (const float2*)keys;
  float2* dst = (float2*)out_keys;
  long n = (long)MM * KK / 2;
  for (long i = (long)blockIdx.x * blockDim.x + threadIdx.x; i < n;
       i += (long)gridDim.x * blockDim.x)
    dst[i] = src[i];
}

__global__ void copy_va(const int* __restrict__ values, const float* __restrict__ age,
                        float* out_values, float* out_age) {
  for (long i = (long)blockIdx.x * blockDim.x + threadIdx.x; i < MM;
       i += (long)gridDim.x * blockDim.x) {
    out_values[i] = (float)values[i];
    out_age[i] = age[i] + 1.0f;    // age = age + 1
  }
}

// --- Memory updates: correct predictions strengthen their key ---
__global__ __launch_bounds__(128) void apply_corr(
    const float* __restrict__ keys, const float* ws_query, const int* ws_result,
    const int* ws_yhatidx, float* out_keys, float* out_age) {
  __shared__ float red[128];
  int b = blockIdx.x, t = threadIdx.x;
  if (!ws_result[b]) return;
  long idx = ws_yhatidx[b];
  float v = keys[idx * KK + t] + ws_query[b * KK + t];
  red[t] = v * v; __syncthreads();
  for (int s = 64; s > 0; s >>= 1) { if (t < s) red[t] += red[t + s]; __syncthreads(); }
  float nrm = fmaxf(sqrtf(red[0]), 1e-12f);
  out_keys[idx * KK + t] = v / nrm;
  if (t == 0) out_age[idx] = 0.f;
}

// --- top-16 oldest (noisy age, post-correction) ---
__global__ __launch_bounds__(512) void oldest_topk(
    const float* __restrict__ out_age, const float* __restrict__ noise, int* ws_oldest) {
  __shared__ float sv[512];
  __shared__ int sm[512];
  __shared__ int stid[512];
  int t = threadIdx.x;
  float lv[16]; int li[16];
#pragma unroll
  for (int i = 0; i < 16; ++i) { lv[i] = -3.4e38f; li[i] = 0; }
  for (long i = t; i < MM; i += 512) {
    float v = out_age[i] + (noise[i] * 2.f - 1.f) * 8.0f;
    if (v > lv[15]) {
      int p = 15;
      while (p > 0 && lv[p - 1] < v) { lv[p] = lv[p - 1]; li[p] = li[p - 1]; --p; }
      lv[p] = v; li[p] = (int)i;
    }
  }
  int hp = 0;
  for (int r = 0; r < 16; ++r) {
    sv[t] = (hp < 16) ? lv[hp] : -3.4e38f;
    sm[t] = (hp < 16) ? li[hp] : 0;
    stid[t] = t;
    __syncthreads();
    for (int s = 256; s > 0; s >>= 1) {
      if (t < s && sv[t + s] > sv[t]) {
        sv[t] = sv[t + s]; sm[t] = sm[t + s]; stid[t] = stid[t + s];
      }
      __syncthreads();
    }
    if (t == 0) ws_oldest[r] = sm[0];
    int w = stid[0];
    __syncthreads();
    if (t == w) ++hp;
  }
}

// --- incorrect predictions overwrite oldest slots ---
__global__ __launch_bounds__(128) void apply_inc(
    const float* ws_query, const int* ws_result, const int* ws_rank,
    const int* ws_oldest, const int* ws_y,
    float* out_keys, float* out_values, float* out_age) {
  int b = blockIdx.x, t = threadIdx.x;
  if (ws_result[b]) return;
  long dest = ws_oldest[ws_rank[b]];
  out_keys[dest * KK + t] = ws_query[b * KK + t];
  if (t == 0) { out_values[dest] = (float)ws_y[b]; out_age[dest] = 0.f; }
}

extern "C" void kernel_launch(void* const* d_in, const int* in_sizes, int n_in,
                              void* d_out, int out_size, void* d_ws, size_t ws_size,
                              hipStream_t stream) {
  const float* loc_preds   = (const float*)d_in[0];
  const float* cls_preds   = (const float*)d_in[1];
  const float* loc_targets = (const float*)d_in[2];
  const int*   cls_targets = (const int*)d_in[3];
  const float* keys        = (const float*)d_in[4];
  const int*   values      = (const int*)d_in[5];
  const float* age         = (const float*)d_in[6];
  const float* noise       = (const float*)d_in[7];
  float* out = (float*)d_out;
  char* ws = (char*)d_ws;

  if (ws_size < (size_t)WS_SCORES + (size_t)BB * MM * sizeof(float)) return;

  float* s_samples = (float*)(ws + WS_SAMPLES);
  float* s_numpos  = (float*)(ws + WS_NUMPOS);
  float* s_locsum  = (float*)(ws + WS_LOCSUM);
  int*   s_y       = (int*)  (ws + WS_Y);
  float* s_query   = (float*)(ws + WS_QUERY);
  int*   s_result  = (int*)  (ws + WS_RESULT);
  int*   s_yhatidx = (int*)  (ws + WS_YHATIDX);
  int*   s_rank    = (int*)  (ws + WS_RANK);
  int*   s_oldest  = (int*)  (ws + WS_OLDEST);
  float* s_cossim  = (float*)(ws + WS_COSSIM);
  int*   s_topk    = (int*)  (ws + WS_TOPKIDX);
  float* s_scores  = (float*)(ws + WS_SCORES);

  float* out_query  = out + OFF_QUERY;
  float* out_keys   = out + OFF_KEYS;
  float* out_values = out + OFF_VALUES;
  float* out_age    = out + OFF_AGE;

  init_ws<<<8, 256, 0, stream>>>(s_samples, s_numpos, s_locsum, s_y);
  reduce_anchors<<<dim3(AA / 128, BB), 128, 0, stream>>>(
      loc_preds, loc_targets, cls_preds, cls_targets,
      s_samples, s_numpos, s_locsum, s_y);
  compute_query<<<BB, 128, 0, stream>>>(s_samples, s_numpos, s_query, out_query);
  scores_wmma<<<MM / 64, 128, 0, stream>>>(keys, s_query, s_scores);
  topk_row<<<BB, 512, 0, stream>>>(s_scores, s_cossim, s_topk);
  finalize_kernel<<<1, 256, 0, stream>>>(
      s_cossim, s_topk, values, s_y, s_numpos, s_locsum,
      s_result, s_yhatidx, s_rank, out);
  copy_keys<<<4096, 256, 0, stream>>>(keys, out_keys);
  copy_va<<<512, 256, 0, stream>>>(values, age, out_values, out_age);
  apply_corr<<<BB, 128, 0, stream>>>(keys, s_query, s_result, s_yhatidx, out_keys, out_age);
  oldest_topk<<<1, 512, 0, stream>>>(out_age, noise, s_oldest);
  apply_inc<<<BB, 128, 0, stream>>>(
      s_query, s_result, s_rank, s_oldest, s_y, out_keys, out_values, out_age);
}